// SlideAttention_65257733096105
// MI455X (gfx1250) — compile-verified
//
#include <hip/hip_runtime.h>
#include <hip/hip_bf16.h>
#include <math.h>
#include <stdint.h>

// ---------------------------------------------------------------------------
// SlideAttention for MI455X (gfx1250, wave32, WMMA + async-to-LDS).
//   K-A:  qkv = x @ qkv_w + [q_bias, 0, v_bias]   (WMMA f32 16x16x4, LDS A-tile)
//   K-B:  fused deformable depthwise + cosine attn (async LDS + wave32 shuffles)
//   K-C:  out = attn_out @ proj_w + proj_b        (WMMA f32 16x16x4, LDS A-tile)
// ---------------------------------------------------------------------------

typedef __attribute__((ext_vector_type(2))) float v2f;
typedef __attribute__((ext_vector_type(4))) float v4f;
typedef __attribute__((ext_vector_type(8))) float v8f;
typedef __attribute__((ext_vector_type(4))) int   int4v;

#define HEADS_ 6
#define CCH    192
#define NQKV   576
#define NPOS   9
#define EPS_   1.55e-05f
#define KDIM   192          // K for both GEMMs
#define SA_STRIDE 196       // padded LDS row stride: 196 mod 64 = 4 -> conflict-free

#if __has_builtin(__builtin_amdgcn_global_load_async_to_lds_b128)
#define HAVE_ASYNC_LDS 1
#else
#define HAVE_ASYNC_LDS 0
#endif

// async DMA global->LDS helpers (ASYNCcnt-tracked; no VGPR round trip)
__device__ __forceinline__ void async_copy16(const float* gsrc, float* ldst) {
#if HAVE_ASYNC_LDS
  __builtin_amdgcn_global_load_async_to_lds_b128(
      (__attribute__((address_space(1))) int4v*)(uintptr_t)gsrc,
      (__attribute__((address_space(3))) int4v*)(uintptr_t)ldst, 0, 0);
#else
  *(v4f*)ldst = *(const v4f*)gsrc;
#endif
}

__device__ __forceinline__ void async_copy4(const float* gsrc, float* ldst) {
#if HAVE_ASYNC_LDS
  __builtin_amdgcn_global_load_async_to_lds_b32(
      (__attribute__((address_space(1))) int*)(uintptr_t)gsrc,
      (__attribute__((address_space(3))) int*)(uintptr_t)ldst, 0, 0);
#else
  *ldst = *gsrc;
#endif
}

__device__ __forceinline__ void async_drain() {
#if HAVE_ASYNC_LDS
  asm volatile("s_wait_asynccnt 0x0" ::: "memory");
#endif
}

// ------------------------- WMMA fp32 GEMM ----------------------------------
// D[M,N] = A[M,K=192] * W[192,N] + bias[N].  Row-major everywhere.
// One block = one 16-row M tile; NSTRIPS waves each own a 16x64 N strip.
// The 16x192 A panel is staged once into LDS (async DMA), then each wave
// reads its A fragment via ds_load_b64.
// Fragment layouts per CDNA5 ISA 7.12.2:
//   A 16x4  : lane m=lane&15; lane<16 -> K={k0,k0+1}, lane>=16 -> K={k0+2,k0+3}
//   B 4x16  : lane n=lane&15; same K split as A
//   C/D     : 8 VGPRs, row M = m0 + r + 8*(lane>=16), col N = n0 + (lane&15)
template <int NSTRIPS>
__global__ __launch_bounds__(NSTRIPS * 32) void gemm_wmma_f32_lds(
    const float* __restrict__ A, const float* __restrict__ W,
    const float* __restrict__ bias, float* __restrict__ Dst,
    int M, int N) {
  __shared__ __attribute__((aligned(16))) float sA[16 * SA_STRIDE];

  const int m0 = blockIdx.x * 16;
  const int tid = threadIdx.x;

  // ---- stage A tile (16 x 192 fp32 = 12KB) into LDS ----
  const int nChunks = 16 * (KDIM / 4);                 // 768 x 16B chunks
  for (int i = tid; i < nChunks; i += NSTRIPS * 32) {
    const int r  = i / (KDIM / 4);
    const int kc = (i % (KDIM / 4)) * 4;
    async_copy16(A + (size_t)(m0 + r) * KDIM + kc, &sA[r * SA_STRIDE + kc]);
  }
  async_drain();                                       // this wave's DMA done
  __syncthreads();                                     // all waves' DMA done

  const int wave = tid >> 5;
  const int lane = tid & 31;
  const int n0 = wave * 64;
  if (n0 >= N || m0 >= M) return;

  const int kHalf = (lane >> 4) << 1;                  // 0 or 2
  const int nCol = lane & 15;
  const float* aBase = &sA[(lane & 15) * SA_STRIDE];

  v8f acc0 = {}, acc1 = {}, acc2 = {}, acc3 = {};

  for (int k0 = 0; k0 < KDIM; k0 += 4) {
    v2f a = *(const v2f*)(aBase + k0 + kHalf);         // ds_load_b64
    const float* w0 = W + (size_t)(k0 + kHalf) * N + n0 + nCol;
    const float* w1 = w0 + N;
    v2f b0; b0.x = w0[0];  b0.y = w1[0];
    v2f b1; b1.x = w0[16]; b1.y = w1[16];
    v2f b2; b2.x = w0[32]; b2.y = w1[32];
    v2f b3; b3.x = w0[48]; b3.y = w1[48];
    acc0 = __builtin_amdgcn_wmma_f32_16x16x4_f32(false, a, false, b0, (short)0, acc0, false, false);
    acc1 = __builtin_amdgcn_wmma_f32_16x16x4_f32(false, a, false, b1, (short)0, acc1, false, false);
    acc2 = __builtin_amdgcn_wmma_f32_16x16x4_f32(false, a, false, b2, (short)0, acc2, false, false);
    acc3 = __builtin_amdgcn_wmma_f32_16x16x4_f32(false, a, false, b3, (short)0, acc3, false, false);
  }

  const int mBase = m0 + ((lane >> 4) << 3);           // +8 for upper half-wave
  v8f accs[4] = {acc0, acc1, acc2, acc3};
#pragma unroll
  for (int t = 0; t < 4; ++t) {
    const int n = n0 + t * 16 + nCol;
    const float bv = bias ? bias[n] : 0.0f;
#pragma unroll
    for (int r = 0; r < 8; ++r) {
      Dst[(size_t)(mBase + r) * N + n] = accs[t][r] + bv;
    }
  }
}

// ------------------------- bias concat -------------------------------------
__global__ void bias_concat_kernel(const float* __restrict__ qb,
                                   const float* __restrict__ vb,
                                   float* __restrict__ out576) {
  int t = blockIdx.x * blockDim.x + threadIdx.x;
  if (t >= NQKV) return;
  float v = 0.0f;
  if (t < CCH) v = qb[t];
  else if (t >= 2 * CCH) v = vb[t - 2 * CCH];
  out576[t] = v;
}

// ------------------------- relative bias MLP -------------------------------
__device__ __forceinline__ float tabval(int d) {
  // d in {-1,0,1}; rel = 4*d; sign(rel)*log1p(|rel|)/log(8)
  if (d == 0) return 0.0f;
  float m = log1pf(4.0f) / logf(8.0f);
  return d > 0 ? m : -m;
}

__global__ void relbias_kernel(const float* __restrict__ w1,   // (2,512)
                               const float* __restrict__ b1,   // (512)
                               const float* __restrict__ w2,   // (512,6)
                               float* __restrict__ rb) {       // (6,9)
  int t = threadIdx.x;
  if (t >= HEADS_ * NPOS) return;
  int h = t / NPOS, j = t % NPOS;
  int i = (NPOS - 1) - j;                 // bias[::-1] row reversal
  float ry = tabval(i / 3 - 1);
  float rx = tabval(i % 3 - 1);
  float acc = 0.0f;
  for (int k = 0; k < 512; ++k) {
    float hid = fmaf(ry, w1[k], fmaf(rx, w1[512 + k], b1[k]));
    hid = fmaxf(hid, 0.0f);
    acc = fmaf(hid, w2[k * HEADS_ + h], acc);
  }
  rb[h * NPOS + j] = 16.0f / (1.0f + __expf(-acc));
}

// ------------------------- fused attention ---------------------------------
__device__ __forceinline__ float wave_sum32(float v) {
#pragma unroll
  for (int m = 16; m >= 1; m >>= 1) v += __shfl_xor(v, m, 32);
  return v;
}

// grid.x = HEADS * (totalPix/8); block = 256 (8 waves); wave = one pixel,
// lane = one of 32 head channels.
__global__ __launch_bounds__(256) void slide_attn_kernel(
    const float* __restrict__ qkv,     // (B*H*W, 576)
    const float* __restrict__ defk,    // (3,3,192,9)
    const float* __restrict__ rb,      // (6,9)
    const float* __restrict__ scale,   // (6)
    float* __restrict__ outbuf,        // (B*H*W, 192)
    int Bb, int Hh, int Ww) {
  __shared__ __attribute__((aligned(16))) float s_defk[NPOS * 32 * NPOS]; // [p][c][j]

  const int head = blockIdx.x % HEADS_;
  const int grp  = blockIdx.x / HEADS_;

  // stage this head's deformable kernel slice into LDS (async DMA, b32 grain)
  for (int i = threadIdx.x; i < NPOS * 32 * NPOS; i += blockDim.x) {
    int j = i % NPOS;
    int c = (i / NPOS) & 31;
    int p = i / (NPOS * 32);
    async_copy4(defk + (size_t)(p * CCH + head * 32 + c) * NPOS + j, &s_defk[i]);
  }
  async_drain();
  __syncthreads();

  const int wave = threadIdx.x >> 5;
  const int lane = threadIdx.x & 31;
  const int pix = grp * 8 + wave;
  const int total = Bb * Hh * Ww;
  if (pix >= total) return;

  const int b = pix / (Hh * Ww);
  const int rrem = pix % (Hh * Ww);
  const int y = rrem / Ww, x = rrem % Ww;
  const int c = lane;

  const float* row = qkv + (size_t)pix * NQKV;
  const float q = row[head * 32 + c];

  // gather 3x3 neighborhood of k/v inputs (zero-padded SAME)
  float kin[NPOS], vin[NPOS];
  bool val[NPOS];
#pragma unroll
  for (int p = 0; p < NPOS; ++p) {
    int yy = y + p / 3 - 1;
    int xx = x + p % 3 - 1;
    bool ok = (yy >= 0) && (yy < Hh) && (xx >= 0) && (xx < Ww);
    val[p] = ok;
    const float* nrow = qkv + (size_t)(((size_t)b * Hh + (ok ? yy : y)) * Ww + (ok ? xx : x)) * NQKV;
    kin[p] = ok ? nrow[CCH + head * 32 + c]     : 0.0f;
    vin[p] = ok ? nrow[2 * CCH + head * 32 + c] : 0.0f;
  }

  // deformable depthwise: out j = sum_p in[p] * (defk[p][c][j] + delta(p==j))
  float kj[NPOS], vj[NPOS];
#pragma unroll
  for (int j = 0; j < NPOS; ++j) {
    float ak = 0.0f, av = 0.0f;
#pragma unroll
    for (int p = 0; p < NPOS; ++p) {
      float w = s_defk[(p * 32 + c) * NPOS + j] + ((p == j) ? 1.0f : 0.0f);
      ak = fmaf(kin[p], w, ak);
      av = fmaf(vin[p], w, av);
    }
    kj[j] = ak;
    vj[j] = av;
  }

  // cosine attention
  float qsq = wave_sum32(q * q);
  float qn = q * rsqrtf(fmaxf(qsq, EPS_)) * __expf(scale[head]);

  float logit[NPOS];
#pragma unroll
  for (int j = 0; j < NPOS; ++j) {
    float ksq = wave_sum32(kj[j] * kj[j]);
    float kn = kj[j] * rsqrtf(fmaxf(ksq, EPS_));
    float dot = wave_sum32(qn * kn);
    logit[j] = dot + (val[j] ? 0.0f : -100.0f) + rb[head * NPOS + j];
  }

  float mx = logit[0];
#pragma unroll
  for (int j = 1; j < NPOS; ++j) mx = fmaxf(mx, logit[j]);
  float den = 0.0f, e[NPOS];
#pragma unroll
  for (int j = 0; j < NPOS; ++j) { e[j] = __expf(logit[j] - mx); den += e[j]; }
  float inv = 1.0f / den;

  float o = 0.0f;
#pragma unroll
  for (int j = 0; j < NPOS; ++j) o = fmaf(e[j] * inv, vj[j], o);

  outbuf[(size_t)pix * CCH + head * 32 + c] = o;
}

// ------------------------- launcher ----------------------------------------
extern "C" void kernel_launch(void* const* d_in, const int* in_sizes, int n_in,
                              void* d_out, int out_size, void* d_ws, size_t ws_size,
                              hipStream_t stream) {
  const float* x       = (const float*)d_in[0];   // (4,96,96,192)
  const float* qkv_w   = (const float*)d_in[1];   // (192,576)
  const float* q_bias  = (const float*)d_in[2];   // (192)
  const float* v_bias  = (const float*)d_in[3];   // (192)
  const float* defk    = (const float*)d_in[4];   // (3,3,192,9)
  const float* scale   = (const float*)d_in[5];   // (6,1,1)
  const float* cpb_w1  = (const float*)d_in[6];   // (2,512)
  const float* cpb_b1  = (const float*)d_in[7];   // (512)
  const float* cpb_w2  = (const float*)d_in[8];   // (512,6)
  const float* proj_w  = (const float*)d_in[9];   // (192,192)
  const float* proj_b  = (const float*)d_in[10];  // (192)
  float* out = (float*)d_out;

  const int B = 4, H = 96, W = 96;
  const int M = B * H * W;                        // 36864 pixels

  // workspace layout (floats): qkv | attn_out | rb | bias576  (~113 MB)
  float* ws       = (float*)d_ws;
  float* qkv      = ws;                                   // M*576
  float* attn_out = qkv + (size_t)M * NQKV;               // M*192
  float* rb       = attn_out + (size_t)M * CCH;           // 64
  float* bias576  = rb + 64;                              // 576

  // small prep kernels
  bias_concat_kernel<<<(NQKV + 255) / 256, 256, 0, stream>>>(q_bias, v_bias, bias576);
  relbias_kernel<<<1, 64, 0, stream>>>(cpb_w1, cpb_b1, cpb_w2, rb);

  // QKV projection: M x 576 = (M x 192) @ (192 x 576); 9 strips -> 288 threads
  gemm_wmma_f32_lds<9><<<M / 16, 9 * 32, 0, stream>>>(x, qkv_w, bias576, qkv, M, NQKV);

  // fused deformable depthwise + cosine attention
  slide_attn_kernel<<<HEADS_ * (M / 8), 256, 0, stream>>>(
      qkv, defk, rb, scale, attn_out, B, H, W);

  // output projection: M x 192 = (M x 192) @ (192 x 192); 3 strips -> 96 threads
  gemm_wmma_f32_lds<3><<<M / 16, 3 * 32, 0, stream>>>(attn_out, proj_w, proj_b, out, M, CCH);
}